// MultiheadAttention_34333968564299
// MI455X (gfx1250) — compile-verified
//
#include <hip/hip_runtime.h>
#include <hip/hip_bf16.h>

// MI455X (gfx1250) fused multihead-attention with top-k sparsification.
// All GEMMs use V_WMMA_F32_16X16X4_F32 (full-precision fp32 matrix pipe).

typedef __attribute__((ext_vector_type(2))) float v2f;
typedef __attribute__((ext_vector_type(8))) float v8f;

#define BB 8
#define SS 1024
#define DD 512
#define HH 8
#define HD 64
#define TOPK 10
#define QKVLD (3 * DD)   // 1536, row stride of qkv buffer
#define SCLD  1028       // padded score-row stride (words): lanes hit disjoint banks
#define VSLD  80         // padded LDS V-tile row stride (words)

__device__ __forceinline__ v8f wmma4(v2f a, v2f b, v8f c) {
  // D = A(16x4) * B(4x16) + C(16x16), fp32
  return __builtin_amdgcn_wmma_f32_16x16x4_f32(
      /*neg_a=*/false, a, /*neg_b=*/false, b,
      /*c_mod=*/(short)0, c, /*reuse_a=*/false, /*reuse_b=*/false);
}

// ---------------------------------------------------------------------------
// C[M,N] = A[M,K] @ W[N,K]^T + bias[N]
// block = 256 threads = 8 waves arranged 2(M) x 4(N); block tile 64 x 128;
// wave tile 32x32 = 2x2 register blocking -> 4 independent WMMA chains.
// ---------------------------------------------------------------------------
__global__ void gemm_xwT_wmma(const float* __restrict__ A,
                              const float* __restrict__ W,
                              const float* __restrict__ bias,
                              float* __restrict__ C,
                              int M, int N, int K) {
  const int lane = threadIdx.x & 31;
  const int wave = threadIdx.x >> 5;
  const int half = lane >> 4;    // K-pair {0,1} vs {2,3}; M row-group 0..7 vs 8..15
  const int l16  = lane & 15;

  const int m0 = blockIdx.y * 64 + (wave >> 2) * 32;
  const int n0 = blockIdx.x * 128 + (wave & 3) * 32;
  if (m0 >= M || n0 >= N) return;

  const float* Ar0 = A + (size_t)(m0 + l16) * K;        // A frag rows (M = lane%16)
  const float* Ar1 = Ar0 + (size_t)16 * K;
  const float* Wr0 = W + (size_t)(n0 + l16) * K;        // B frag cols (N = lane%16)
  const float* Wr1 = Wr0 + (size_t)16 * K;

  v8f acc00 = {}, acc01 = {}, acc10 = {}, acc11 = {};
#pragma unroll 4
  for (int k = 0; k < K; k += 4) {
    if ((k & 63) == 0) {
      __builtin_prefetch(Ar0 + k + 64, 0, 3);   // global_prefetch_b8 (near)
      __builtin_prefetch(Wr0 + k + 64, 0, 3);
    }
    const int ko = k + half * 2;
    v2f a0 = *(const v2f*)(Ar0 + ko);
    v2f a1 = *(const v2f*)(Ar1 + ko);
    v2f b0 = *(const v2f*)(Wr0 + ko);
    v2f b1 = *(const v2f*)(Wr1 + ko);
    acc00 = wmma4(a0, b0, acc00);
    acc01 = wmma4(a0, b1, acc01);
    acc10 = wmma4(a1, b0, acc10);
    acc11 = wmma4(a1, b1, acc11);
  }

  const float bv0 = bias[n0 + l16];
  const float bv1 = bias[n0 + 16 + l16];
#pragma unroll
  for (int r = 0; r < 8; ++r) {
    const size_t row0 = (size_t)(m0 + half * 8 + r) * N;
    const size_t row1 = (size_t)(m0 + 16 + half * 8 + r) * N;
    C[row0 + n0 + l16]      = acc00[r] + bv0;
    C[row0 + n0 + 16 + l16] = acc01[r] + bv1;
    C[row1 + n0 + l16]      = acc10[r] + bv0;
    C[row1 + n0 + 16 + l16] = acc11[r] + bv1;
  }
}

// ---------------------------------------------------------------------------
// Fused attention: per block handles one (b,h) and a 16-query strip.
//   phase 1: logits(16x1024) = Q(16x64) K^T / sqrt(64)  -> LDS (dual-acc)
//   phase 2: softmax + top-10 threshold / rest-mean replacement (wave32)
//   phase 3: O(16x64) = P(16x1024) V(1024x64); V staged in LDS chunks,
//            all 8 waves (4 N-tiles x 2 K-halves), partials combined via LDS
// dynamic LDS: sc[16][SCLD] + Vs[128][VSLD] + topv[16][10] + topi[16][10]
// ---------------------------------------------------------------------------
__global__ void attn_topk_wmma(const float* __restrict__ qkv,
                               float* __restrict__ attn_out) {
  extern __shared__ float smem[];
  float* sc   = smem;                            // 16*SCLD scores
  float* Vs   = smem + 16 * SCLD;                // 128*VSLD staged V chunk
  float* topv = Vs + 128 * VSLD;                 // 16*10 extracted values
  int*   topi = (int*)(topv + 16 * TOPK);        // 16*10 extracted indices
  float* partial = Vs;                           // reuse Vs after last chunk

  const int lane = threadIdx.x & 31;
  const int wave = threadIdx.x >> 5;             // 0..7
  const int half = lane >> 4;
  const int l16  = lane & 15;

  const int bh = blockIdx.y;                     // 0..63
  const int b  = bh >> 3;
  const int h  = bh & (HH - 1);
  const int q0 = blockIdx.x * 16;

  const float* Qb = qkv + (size_t)b * SS * QKVLD + h * (3 * HD);   // +0   : Q
  const float* Kb = Qb + HD;                                       // +64  : K
  const float* Vb = Qb + 2 * HD;                                   // +128 : V

  // ---- phase 1: logits -> LDS ----
  v2f qa[16];  // hoisted Q fragments for all 16 k-steps (HD = 16*4)
  {
    const float* qrow = Qb + (size_t)(q0 + l16) * QKVLD;
#pragma unroll
    for (int ks = 0; ks < 16; ++ks)
      qa[ks] = *(const v2f*)(qrow + ks * 4 + half * 2);
  }
  const float scale = 0.125f;  // 1/sqrt(64)
  for (int p = 0; p < 4; ++p) {
    const int n0a = (wave + p * 16) * 16;
    const int n0b = (wave + 8 + p * 16) * 16;
    const float* kr0 = Kb + (size_t)(n0a + l16) * QKVLD;
    const float* kr1 = Kb + (size_t)(n0b + l16) * QKVLD;
    v8f acc0 = {}, acc1 = {};
#pragma unroll
    for (int ks = 0; ks < 16; ++ks) {
      v2f bf0 = *(const v2f*)(kr0 + ks * 4 + half * 2);
      v2f bf1 = *(const v2f*)(kr1 + ks * 4 + half * 2);
      acc0 = wmma4(qa[ks], bf0, acc0);
      acc1 = wmma4(qa[ks], bf1, acc1);
    }
#pragma unroll
    for (int r = 0; r < 8; ++r) {
      sc[(half * 8 + r) * SCLD + n0a + l16] = acc0[r] * scale;
      sc[(half * 8 + r) * SCLD + n0b + l16] = acc1[r] * scale;
    }
  }
  __syncthreads();

  // ---- phase 2: softmax + top-k (one wave per row, 2 rows per wave) ----
  for (int rr = 0; rr < 2; ++rr) {
    const int row = wave + rr * 8;
    float* p = sc + row * SCLD;

    float mx = -3.4e38f;
    for (int j = 0; j < 32; ++j) mx = fmaxf(mx, p[lane + j * 32]);
    for (int o = 16; o > 0; o >>= 1) mx = fmaxf(mx, __shfl_xor(mx, o, 32));

    float sum = 0.f;                 // unnormalized exp; normalize at the end
    for (int j = 0; j < 32; ++j) {
      const int c = lane + j * 32;
      float e = __expf(p[c] - mx);
      p[c] = e;
      sum += e;
    }
    for (int o = 16; o > 0; o >>= 1) sum += __shfl_xor(sum, o, 32);

    float topsum = 0.f, thresh = 0.f;
    for (int t = 0; t < TOPK; ++t) {
      float lm = -1.f;
      int   li = -1;
      for (int j = 0; j < 32; ++j) {
        const int c = lane + j * 32;
        const float v = p[c];
        if (v > lm) { lm = v; li = c; }
      }
      for (int o = 16; o > 0; o >>= 1) {
        const float ov = __shfl_xor(lm, o, 32);
        const int   oi = __shfl_xor(li, o, 32);
        if (ov > lm || (ov == lm && oi < li)) { lm = ov; li = oi; }
      }
      topsum += lm;
      thresh = lm;                   // after loop: 10th largest
      if ((li & 31) == lane) {       // winning lane removes + records
        p[li] = -1.f;
        topv[row * TOPK + t] = lm;
        topi[row * TOPK + t] = li;
      }
    }
    const float restmean = (sum - topsum) / (float)(SS - TOPK);

    if (lane < TOPK) p[topi[row * TOPK + lane]] = topv[row * TOPK + lane];

    const float inv = 1.f / sum;
    for (int j = 0; j < 32; ++j) {
      const int c = lane + j * 32;
      const float v = p[c];
      p[c] = (v >= thresh ? v : restmean) * inv;
    }
  }
  __syncthreads();

  // ---- phase 3: O = P @ V with LDS-staged V chunks ----
  const int kt = wave >> 2;            // K-half within chunk (0/1)
  const int n0 = (wave & 3) * 16;      // N-tile within HD
  v8f acc = {};
  for (int chunk = 0; chunk < SS; chunk += 128) {
    // cooperative coalesced stage: V[chunk..chunk+127][0..63] -> Vs
#pragma unroll
    for (int i = 0; i < 8; ++i) {
      const int idx4 = threadIdx.x + i * 256;        // 0..2047 float4 slots
      const int r  = idx4 >> 4;                      // V row 0..127
      const int c4 = idx4 & 15;                      // float4 col
      *(float4*)&Vs[r * VSLD + c4 * 4] =
          *(const float4*)&Vb[(size_t)(chunk + r) * QKVLD + c4 * 4];
    }
    __syncthreads();
    const int kbase = kt * 64;
#pragma unroll 8
    for (int kk = kbase; kk < kbase + 64; kk += 4) {
      v2f a = *(const v2f*)(sc + l16 * SCLD + chunk + kk + half * 2);
      v2f bf;
      bf[0] = Vs[(kk + half * 2 + 0) * VSLD + n0 + l16];
      bf[1] = Vs[(kk + half * 2 + 1) * VSLD + n0 + l16];
      acc = wmma4(a, bf, acc);
    }
    __syncthreads();
  }

  // combine the two K-halves via LDS (reuses Vs space)
  if (kt == 1) {
#pragma unroll
    for (int r = 0; r < 8; ++r)
      partial[(wave & 3) * 256 + (half * 8 + r) * 16 + l16] = acc[r];
  }
  __syncthreads();
  if (kt == 0) {
#pragma unroll
    for (int r = 0; r < 8; ++r) {
      const float v = acc[r] + partial[(wave & 3) * 256 + (half * 8 + r) * 16 + l16];
      const int qr = q0 + half * 8 + r;
      attn_out[((size_t)b * SS + qr) * DD + h * HD + n0 + l16] = v;
    }
  }
}

// ---------------------------------------------------------------------------
extern "C" void kernel_launch(void* const* d_in, const int* in_sizes, int n_in,
                              void* d_out, int out_size, void* d_ws, size_t ws_size,
                              hipStream_t stream) {
  const float* x    = (const float*)d_in[0];   // (B,S,D)
  const float* Wqkv = (const float*)d_in[1];   // (3D,D)
  const float* bqkv = (const float*)d_in[2];   // (3D,)
  const float* Wo   = (const float*)d_in[3];   // (D,D)
  const float* bo   = (const float*)d_in[4];   // (D,)
  float* out = (float*)d_out;                  // (B,S,D)

  float* qkv      = (float*)d_ws;                                  // B*S*3D f32 = 48 MB
  float* attn_out = qkv + (size_t)BB * SS * QKVLD;                 // B*S*D  f32 = 16 MB

  const int M = BB * SS;  // 8192

  // 1) QKV projection
  {
    dim3 grid(QKVLD / 128, M / 64);
    gemm_xwT_wmma<<<grid, 256, 0, stream>>>(x, Wqkv, bqkv, qkv, M, QKVLD, DD);
  }

  // 2) fused attention + topk
  {
    dim3 grid(SS / 16, BB * HH);
    const size_t smem = (size_t)(16 * SCLD + 128 * VSLD) * sizeof(float)
                      + (size_t)(16 * TOPK) * (sizeof(float) + sizeof(int));
    attn_topk_wmma<<<grid, 256, smem, stream>>>(qkv, attn_out);
  }

  // 3) output projection
  {
    dim3 grid(DD / 128, M / 64);
    gemm_xwT_wmma<<<grid, 256, 0, stream>>>(attn_out, Wo, bo, out, M, DD, DD);
  }
}